// GCN_83829171684013
// MI455X (gfx1250) — compile-verified
//
#include <hip/hip_runtime.h>
#include <hip/hip_bf16.h>
#include <math.h>

// ---------------------------------------------------------------------------
// 2-layer GCN forward for MI455X (gfx1250, wave32).
// Dense GEMMs use V_WMMA_F32_16X16X4_F32 (fp32 end-to-end, matches reference).
// Scatter/gather message passing uses L2-resident fp32 atomics.
// ---------------------------------------------------------------------------

typedef __attribute__((ext_vector_type(2))) float v2f;
typedef __attribute__((ext_vector_type(8))) float v8f;

#define NNODES   100000
#define NFEAT    512
#define NHID     16
#define NCLS     40
#define NEDGES   3200000

static __device__ __forceinline__ v8f wmma_f32_16x16x4(v2f a, v2f b, v8f c) {
  return __builtin_amdgcn_wmma_f32_16x16x4_f32(
      /*neg_a=*/false, a, /*neg_b=*/false, b,
      /*c_mod=*/(short)0, c, /*reuse_a=*/false, /*reuse_b=*/false);
}

// ---------------------------- degree / norm --------------------------------

__global__ void k_deg_init(unsigned* __restrict__ deg) {
  int i = blockIdx.x * blockDim.x + threadIdx.x;
  if (i < NNODES) deg[i] = 1u;                 // self-loop
}

__global__ void k_deg_count(const int* __restrict__ dst,
                            unsigned* __restrict__ deg) {
  int e = blockIdx.x * blockDim.x + threadIdx.x;
  if (e < NEDGES) atomicAdd(&deg[dst[e]], 1u);
}

__global__ void k_dinv(const unsigned* __restrict__ deg,
                       float* __restrict__ dinv) {
  int i = blockIdx.x * blockDim.x + threadIdx.x;
  if (i < NNODES) dinv[i] = rsqrtf((float)deg[i]);   // deg >= 1 always
}

// ------------------------- GEMM1: H1 = X @ W1 ------------------------------
// One wave = one 16-row tile; K loop in steps of 4 via WMMA f32 16x16x4.
// A layout: lane l holds row (l&15); VGPR0/1 = K {0,1} (lanes 0-15) or {2,3}
// (lanes 16-31) -> contiguous float2 load from the row.
// B layout: lane l holds col (l&15) of K-row {0,1} / {2,3} per half-wave.
// C/D: VGPR j = rows j and j+8, col = lane&15.

__global__ __launch_bounds__(256) void k_gemm1(const float* __restrict__ X,
                                               const float* __restrict__ W1,
                                               float* __restrict__ H1) {
  __shared__ float sW[NFEAT * NHID];                 // 32 KB of 320 KB LDS
  for (int i = threadIdx.x; i < NFEAT * NHID; i += 256) sW[i] = W1[i];
  __syncthreads();

  const int wave = threadIdx.x >> 5;
  const int lane = threadIdx.x & 31;
  const int tile = blockIdx.x * 8 + wave;
  if (tile * 16 >= NNODES) return;                   // wave-uniform: EXEC stays full

  const int m    = lane & 15;
  const int koff = (lane >> 4) << 1;                 // 0 or 2
  const float* arow = X + (size_t)(tile * 16 + m) * NFEAT;

  v8f acc = {};
  for (int k0 = 0; k0 < NFEAT; k0 += 4) {
    v2f a = *(const v2f*)(arow + k0 + koff);         // 8B-aligned (k0+koff even)
    v2f b;
    b.x = sW[(k0 + koff) * NHID + m];
    b.y = sW[(k0 + koff + 1) * NHID + m];
    acc = wmma_f32_16x16x4(a, b, acc);
  }

  const int mbase = (lane < 16) ? 0 : 8;
  float* out = H1 + (size_t)(tile * 16 + mbase) * NHID + m;
#pragma unroll
  for (int j = 0; j < 8; ++j) out[j * NHID] = acc[j];
}

// ---------------- layer-1 aggregation (self-loop init + edges) -------------

__global__ void k_init1(const float* __restrict__ H1,
                        const float* __restrict__ dinv,
                        float* __restrict__ A1) {
  int idx = blockIdx.x * blockDim.x + threadIdx.x;   // over NNODES*16
  if (idx >= NNODES * NHID) return;
  int i = idx >> 4;
  float w = dinv[i] * dinv[i];
  A1[idx] = H1[idx] * w;
}

__global__ void k_scatter1(const int* __restrict__ src,
                           const int* __restrict__ dst,
                           const float* __restrict__ dinv,
                           const float* __restrict__ H1,
                           float* __restrict__ A1) {
  int idx = blockIdx.x * blockDim.x + threadIdx.x;   // over NEDGES*16
  if (idx >= NEDGES * NHID) return;
  int e = idx >> 4, f = idx & 15;
  int s = src[e], d = dst[e];
  float w = dinv[s] * dinv[d];
  atomicAdd(&A1[(size_t)d * NHID + f], H1[(size_t)s * NHID + f] * w);
}

// --------------- GEMM2: H2 = relu(A1 + b1) @ W2  (N = 40 -> 3 tiles) -------

__global__ __launch_bounds__(256) void k_gemm2(const float* __restrict__ A1,
                                               const float* __restrict__ b1,
                                               const float* __restrict__ W2,
                                               float* __restrict__ H2) {
  const int wave = threadIdx.x >> 5;
  const int lane = threadIdx.x & 31;
  const int tile = blockIdx.x * 8 + wave;
  if (tile * 16 >= NNODES) return;

  const int m    = lane & 15;
  const int koff = (lane >> 4) << 1;
  const float* arow = A1 + (size_t)(tile * 16 + m) * NHID;

  v8f acc0 = {}, acc1 = {}, acc2 = {};
#pragma unroll
  for (int kt = 0; kt < 4; ++kt) {
    const int kk = kt * 4 + koff;                    // 0..15
    v2f a;
    a.x = fmaxf(arow[kk] + b1[kk], 0.0f);            // fused bias + ReLU
    a.y = fmaxf(arow[kk + 1] + b1[kk + 1], 0.0f);
    v2f b0, b1v, b2v;
    b0.x  = W2[kk * NCLS + m];        b0.y  = W2[(kk + 1) * NCLS + m];
    b1v.x = W2[kk * NCLS + 16 + m];   b1v.y = W2[(kk + 1) * NCLS + 16 + m];
    const int c2 = 32 + m;
    b2v.x = (c2 < NCLS) ? W2[kk * NCLS + c2] : 0.0f;
    b2v.y = (c2 < NCLS) ? W2[(kk + 1) * NCLS + c2] : 0.0f;
    acc0 = wmma_f32_16x16x4(a, b0, acc0);
    acc1 = wmma_f32_16x16x4(a, b1v, acc1);
    acc2 = wmma_f32_16x16x4(a, b2v, acc2);
  }

  const int mbase = (lane < 16) ? 0 : 8;
#pragma unroll
  for (int j = 0; j < 8; ++j) {
    const size_t node = (size_t)(tile * 16 + mbase + j);
    H2[node * NCLS + m]       = acc0[j];
    H2[node * NCLS + 16 + m]  = acc1[j];
    if (32 + m < NCLS) H2[node * NCLS + 32 + m] = acc2[j];
  }
}

// ---------------- layer-2 aggregation (into d_out) -------------------------

__global__ void k_init2(const float* __restrict__ H2,
                        const float* __restrict__ dinv,
                        float* __restrict__ OUT) {
  int idx = blockIdx.x * blockDim.x + threadIdx.x;   // over NNODES*40
  if (idx >= NNODES * NCLS) return;
  int i = idx / NCLS;
  float w = dinv[i] * dinv[i];
  OUT[idx] = H2[idx] * w;
}

__global__ void k_scatter2(const int* __restrict__ src,
                           const int* __restrict__ dst,
                           const float* __restrict__ dinv,
                           const float* __restrict__ H2,
                           float* __restrict__ OUT) {
  int idx = blockIdx.x * blockDim.x + threadIdx.x;   // over NEDGES*40 = 128M
  if (idx >= NEDGES * NCLS) return;
  int e = idx / NCLS, c = idx - e * NCLS;
  int s = src[e], d = dst[e];
  float w = dinv[s] * dinv[d];
  atomicAdd(&OUT[(size_t)d * NCLS + c], H2[(size_t)s * NCLS + c] * w);
}

// ---------------- bias2 + log_softmax (in place on d_out) ------------------

__global__ void k_lsm(float* __restrict__ OUT, const float* __restrict__ b2) {
  int i = blockIdx.x * blockDim.x + threadIdx.x;
  if (i >= NNODES) return;
  float* row = OUT + (size_t)i * NCLS;
  float mx = -INFINITY;
#pragma unroll 8
  for (int c = 0; c < NCLS; ++c) mx = fmaxf(mx, row[c] + b2[c]);
  float s = 0.0f;
#pragma unroll 8
  for (int c = 0; c < NCLS; ++c) s += __expf(row[c] + b2[c] - mx);
  const float l = __logf(s);
#pragma unroll 8
  for (int c = 0; c < NCLS; ++c) row[c] = row[c] + b2[c] - mx - l;
}

// ---------------------------------------------------------------------------

extern "C" void kernel_launch(void* const* d_in, const int* in_sizes, int n_in,
                              void* d_out, int out_size, void* d_ws, size_t ws_size,
                              hipStream_t stream) {
  (void)in_sizes; (void)n_in; (void)out_size; (void)ws_size;

  const float* X  = (const float*)d_in[0];
  const float* W1 = (const float*)d_in[1];
  const float* b1 = (const float*)d_in[2];
  const float* W2 = (const float*)d_in[3];
  const float* b2 = (const float*)d_in[4];
  const int*   ei = (const int*)d_in[5];   // JAX x64-off: int64 request -> int32
  float* OUT = (float*)d_out;

  const int* src = ei;
  const int* dst = ei + NEDGES;

  // workspace carve-up (256B aligned), ~29.7 MB total
  char* ws = (char*)d_ws;
  size_t off = 0;
  auto take = [&](size_t bytes) -> char* {
    char* p = ws + off;
    off = (off + bytes + 255) & ~(size_t)255;
    return p;
  };
  unsigned* deg  = (unsigned*)take((size_t)NNODES * 4);
  float*    dinv = (float*)take((size_t)NNODES * 4);
  float*    H1   = (float*)take((size_t)NNODES * NHID * 4);
  float*    A1   = (float*)take((size_t)NNODES * NHID * 4);
  float*    H2   = (float*)take((size_t)NNODES * NCLS * 4);

  const int T = 256;
  const int gemm_blocks = (NNODES / 16 + 7) / 8;     // 6250 tiles, 8 waves/block

  k_deg_init <<<(NNODES + T - 1) / T, T, 0, stream>>>(deg);
  k_deg_count<<<(NEDGES + T - 1) / T, T, 0, stream>>>(dst, deg);
  k_dinv     <<<(NNODES + T - 1) / T, T, 0, stream>>>(deg, dinv);

  k_gemm1    <<<gemm_blocks, T, 0, stream>>>(X, W1, H1);
  k_init1    <<<(NNODES * NHID + T - 1) / T, T, 0, stream>>>(H1, dinv, A1);
  k_scatter1 <<<(NEDGES * NHID + T - 1) / T, T, 0, stream>>>(src, dst, dinv, H1, A1);

  k_gemm2    <<<gemm_blocks, T, 0, stream>>>(A1, b1, W2, H2);
  k_init2    <<<(NNODES * NCLS + T - 1) / T, T, 0, stream>>>(H2, dinv, OUT);
  k_scatter2 <<<(NEDGES * NCLS + T - 1) / T, T, 0, stream>>>(src, dst, dinv, H2, OUT);

  k_lsm      <<<(NNODES + T - 1) / T, T, 0, stream>>>(OUT, b2);
}